// Net_55645596287043
// MI455X (gfx1250) — compile-verified
//
#include <hip/hip_runtime.h>
#include <math.h>

typedef __attribute__((ext_vector_type(2))) float v2f;
typedef __attribute__((ext_vector_type(8))) float v8f;

#define BN_EPS 1e-5f

// ---------------- workspace layout (floats) ----------------
// part : 1352 blocks * 6 comps (sum[3], sumsq[3])          @ 0
// stats: a[3], d[3] (folded BN affine)                      @ 8192
// Araw : 512*676 sites * 9 raw conv dot-products            @ 8704     (3,115,008)
// A    : 512 x 2028 activations (post conv2+relu)           @ 3123712  (1,038,336)
// B    : 2028 x 128 gathered fc1 weights                    @ 4162048  (  259,584)
// F    : 512 x 128 fc1 activations                          @ 4421632  (   65,536)
// total = 4,487,168 floats ~= 17.9 MB
static const int K_FC1   = 2028;   // 3*26*26 nonzero pooled features
static const int PART_N  = 1352;   // conv grid blocks
static const int ST_OFF  = 8192;
static const int RAW_OFF = 8704;
static const int A_OFF   = RAW_OFF + 512 * 676 * 9;
static const int B_OFF   = A_OFF + 512 * 2028;
static const int F_OFF   = B_OFF + 2028 * 128;

// ---- Kernel 1: single pass over x ----
// Each thread owns one (b,i,j) conv site. The stride-5/pad-1/k5 windows tile
// the 128x128 plane exactly (disjoint + complete), so the same pass yields
// exact BN partial sums: every x element is touched exactly once.
// Outputs: Araw[site*9 + co*3 + ci] = sum over valid taps of w1[co,ci,ky,kx]*x
//          part[block*6 + {c, 3+c}] = per-block {sum, sumsq} per channel
__global__ void conv_raw_stats(const float* __restrict__ x,
                               const float* __restrict__ w1,
                               float* __restrict__ Araw,
                               float* __restrict__ part) {
    __shared__ float sw1[225];
    for (int t = threadIdx.x; t < 225; t += blockDim.x) sw1[t] = w1[t];
    __syncthreads();

    const int tid = blockIdx.x * blockDim.x + threadIdx.x;  // 512*676 = 346112 exact
    const int b   = tid / 676;
    const int rem = tid % 676;
    const int i   = rem / 26;
    const int j   = rem % 26;

    const int r0 = 5 * i - 1;   // pad = 1
    const int c0 = 5 * j - 1;

    float S[9];                 // S[co*3+ci]
    #pragma unroll
    for (int t = 0; t < 9; ++t) S[t] = 0.f;
    float sum[3] = {0.f, 0.f, 0.f};
    float sq[3]  = {0.f, 0.f, 0.f};

    const float* xb = x + (size_t)b * 3 * 16384;
    for (int ci = 0; ci < 3; ++ci) {
        const float* xp  = xb + ci * 16384;
        const float* wci = sw1 + ci * 25;
        #pragma unroll
        for (int ky = 0; ky < 5; ++ky) {
            const int r = r0 + ky;
            if (r < 0 || r > 127) continue;
            const float* xr = xp + r * 128;
            const float* wr = wci + ky * 5;
            #pragma unroll
            for (int kx = 0; kx < 5; ++kx) {
                const int cc = c0 + kx;
                if (cc < 0 || cc > 127) continue;
                const float v = xr[cc];
                sum[ci] += v;
                sq[ci]  = fmaf(v, v, sq[ci]);
                S[0 * 3 + ci] = fmaf(wr[0 * 75 + kx], v, S[0 * 3 + ci]);
                S[1 * 3 + ci] = fmaf(wr[1 * 75 + kx], v, S[1 * 3 + ci]);
                S[2 * 3 + ci] = fmaf(wr[2 * 75 + kx], v, S[2 * 3 + ci]);
            }
        }
    }
    float* Ar = Araw + (size_t)tid * 9;
    #pragma unroll
    for (int t = 0; t < 9; ++t) Ar[t] = S[t];

    // block reduction of the 6 stat components (deterministic tree)
    __shared__ float red[256];
    #pragma unroll
    for (int comp = 0; comp < 6; ++comp) {
        red[threadIdx.x] = (comp < 3) ? sum[comp] : sq[comp - 3];
        __syncthreads();
        for (int o = 128; o > 0; o >>= 1) {
            if (threadIdx.x < o) red[threadIdx.x] += red[threadIdx.x + o];
            __syncthreads();
        }
        if (threadIdx.x == 0) part[(size_t)blockIdx.x * 6 + comp] = red[0];
        __syncthreads();
    }
}

// ---- Kernel 2: deterministic finalize -> a[c], d[c] ----
__global__ void bn_finalize(const float* __restrict__ part,
                            const float* __restrict__ gamma,
                            const float* __restrict__ beta,
                            float* __restrict__ stats) {
    __shared__ float ss[256], sq[256];
    for (int c = 0; c < 3; ++c) {
        float s = 0.f, s2 = 0.f;
        for (int p = threadIdx.x; p < PART_N; p += 256) {
            s  += part[(size_t)p * 6 + c];
            s2 += part[(size_t)p * 6 + 3 + c];
        }
        ss[threadIdx.x] = s; sq[threadIdx.x] = s2;
        __syncthreads();
        for (int o = 128; o > 0; o >>= 1) {
            if (threadIdx.x < o) {
                ss[threadIdx.x] += ss[threadIdx.x + o];
                sq[threadIdx.x] += sq[threadIdx.x + o];
            }
            __syncthreads();
        }
        if (threadIdx.x == 0) {
            const float N = 512.f * 16384.f;
            float mean = ss[0] / N;
            float var  = sq[0] / N - mean * mean;
            float a = gamma[c] * rsqrtf(var + BN_EPS);
            stats[c]     = a;
            stats[3 + c] = beta[c] - mean * a;
        }
        __syncthreads();
    }
}

// ---- Kernel 3: combine raw convs with BN affine, ReLU, 1x1 conv2, ReLU ----
// out1[co] = b1[co] + sum_ci( a_ci * Sraw[co][ci] + d_ci * Wsum[co][ci](i,j) )
// where Wsum is the sum of w1 taps inside the valid (non-padded) window.
// Writes A[b*2028 + co*676 + i*26 + j].
__global__ void conv_combine(const float* __restrict__ Araw,
                             const float* __restrict__ stats,
                             const float* __restrict__ w1, const float* __restrict__ b1,
                             const float* __restrict__ w2, const float* __restrict__ b2,
                             float* __restrict__ A) {
    __shared__ float sw1[225], sb1[3], sw2[9], sb2[3], sa[3], sd[3];
    for (int t = threadIdx.x; t < 225; t += blockDim.x) sw1[t] = w1[t];
    if (threadIdx.x < 3) {
        sb1[threadIdx.x] = b1[threadIdx.x];
        sb2[threadIdx.x] = b2[threadIdx.x];
        sa[threadIdx.x]  = stats[threadIdx.x];
        sd[threadIdx.x]  = stats[3 + threadIdx.x];
    }
    if (threadIdx.x < 9)  // w2[co][ci][0][0] of a (3,3,5,5) tensor
        sw2[threadIdx.x] = w2[(threadIdx.x / 3) * 75 + (threadIdx.x % 3) * 25];
    __syncthreads();

    const int tid = blockIdx.x * blockDim.x + threadIdx.x;  // 512*676 exact
    const int b   = tid / 676;
    const int rem = tid % 676;
    const int i   = rem / 26;
    const int j   = rem % 26;

    const int kylo = (i == 0) ? 1 : 0, kyhi = (i == 25) ? 3 : 4;
    const int kxlo = (j == 0) ? 1 : 0, kxhi = (j == 25) ? 3 : 4;

    const float* Ar = Araw + (size_t)tid * 9;
    float out1[3];
    #pragma unroll
    for (int co = 0; co < 3; ++co) {
        float acc = sb1[co];
        #pragma unroll
        for (int ci = 0; ci < 3; ++ci) {
            float wsum = 0.f;
            const float* wci = sw1 + co * 75 + ci * 25;
            for (int ky = kylo; ky <= kyhi; ++ky)
                for (int kx = kxlo; kx <= kxhi; ++kx)
                    wsum += wci[ky * 5 + kx];
            acc = fmaf(sa[ci], Ar[co * 3 + ci], acc);
            acc = fmaf(sd[ci], wsum, acc);
        }
        out1[co] = fmaxf(acc, 0.f);
    }

    float* Ab = A + (size_t)b * K_FC1 + i * 26 + j;
    #pragma unroll
    for (int co = 0; co < 3; ++co) {
        float o = sb2[co] + sw2[co * 3 + 0] * out1[0]
                          + sw2[co * 3 + 1] * out1[1]
                          + sw2[co * 3 + 2] * out1[2];
        Ab[co * 676] = fmaxf(o, 0.f);
    }
}

// ---- Kernel 4: gather the 2028 active columns of fc1_w -> B[k][n] ----
// feature (c,i,j) lives at flat column c*4225 + floor(5i/2)*65 + floor(5j/2)
__global__ void pack_B(const float* __restrict__ fc1_w, float* __restrict__ Bm) {
    const int n = threadIdx.x;   // 0..127
    const int k = blockIdx.x;    // 0..2027
    const int c = k / 676;
    const int rem = k % 676;
    const int i = rem / 26;
    const int j = rem % 26;
    const int col = c * 4225 + ((5 * i) >> 1) * 65 + ((5 * j) >> 1);
    Bm[(size_t)k * 128 + n] = fc1_w[(size_t)n * 12675 + col];
}

// ---- Kernel 5: fc1 GEMM [512 x 2028] x [2028 x 128] via V_WMMA_F32_16X16X4_F32 ----
// One wave computes one 16x16 tile of F; 507 k-steps. Bias + ReLU fused.
__global__ void gemm_fc1_wmma(const float* __restrict__ A,
                              const float* __restrict__ Bm,
                              const float* __restrict__ fc1_b,
                              float* __restrict__ F) {
    const int waveId = (blockIdx.x * blockDim.x + threadIdx.x) >> 5;  // 0..255
    const int lane   = threadIdx.x & 31;
    const int mt = waveId >> 3;     // 0..31  -> rows mt*16..mt*16+15
    const int nt = waveId & 7;      // 0..7   -> cols nt*16..nt*16+15
    const int half = lane >> 4;     // selects K pair {0,1} vs {2,3}
    const int l16  = lane & 15;     // A: row M; B: col N

    const float* arow = A + (size_t)(mt * 16 + l16) * K_FC1;
    const float* bcol = Bm + nt * 16 + l16;
    v8f c = {};
    for (int k0 = 0; k0 < K_FC1; k0 += 4) {
        const int ka = k0 + 2 * half;
        v2f a, b;
        a.x = arow[ka];
        a.y = arow[ka + 1];
        b.x = bcol[(size_t)ka * 128];
        b.y = bcol[(size_t)(ka + 1) * 128];
        // 8 args: (neg_a, A, neg_b, B, c_mod, C, reuse_a, reuse_b)
        c = __builtin_amdgcn_wmma_f32_16x16x4_f32(false, a, false, b,
                                                  (short)0, c, false, false);
    }
    const float bias = fc1_b[nt * 16 + l16];
    #pragma unroll
    for (int r = 0; r < 8; ++r) {
        const int row = mt * 16 + r + 8 * half;   // C layout: VGPR r, half selects +8
        const float v = c[r] + bias;
        F[(size_t)row * 128 + nt * 16 + l16] = fmaxf(v, 0.f);
    }
}

// ---- Kernel 6: fc2 (128 -> 10) + log_softmax; one thread per batch row ----
__global__ void fc2_logsoftmax(const float* __restrict__ F,
                               const float* __restrict__ fc2_w,
                               const float* __restrict__ fc2_b,
                               float* __restrict__ out) {
    const int b = blockIdx.x * blockDim.x + threadIdx.x;
    if (b >= 512) return;
    const float* f = F + (size_t)b * 128;
    float lg[10];
    float mx = -1e30f;
    #pragma unroll
    for (int o = 0; o < 10; ++o) {
        float s = fc2_b[o];
        const float* wr = fc2_w + o * 128;
        for (int n = 0; n < 128; ++n) s = fmaf(f[n], wr[n], s);
        lg[o] = s;
        mx = fmaxf(mx, s);
    }
    float se = 0.f;
    #pragma unroll
    for (int o = 0; o < 10; ++o) se += expf(lg[o] - mx);
    const float lse = mx + logf(se);
    #pragma unroll
    for (int o = 0; o < 10; ++o) out[(size_t)b * 10 + o] = lg[o] - lse;
}

extern "C" void kernel_launch(void* const* d_in, const int* in_sizes, int n_in,
                              void* d_out, int out_size, void* d_ws, size_t ws_size,
                              hipStream_t stream) {
    const float* x      = (const float*)d_in[0];
    const float* gamma  = (const float*)d_in[1];
    const float* beta   = (const float*)d_in[2];
    const float* w1     = (const float*)d_in[3];
    const float* b1     = (const float*)d_in[4];
    const float* w2     = (const float*)d_in[5];
    const float* b2     = (const float*)d_in[6];
    const float* fc1_w  = (const float*)d_in[7];
    const float* fc1_b  = (const float*)d_in[8];
    const float* fc2_w  = (const float*)d_in[9];
    const float* fc2_b  = (const float*)d_in[10];
    float* out = (float*)d_out;

    float* ws    = (float*)d_ws;
    float* part  = ws;              // 1352 * 6
    float* stats = ws + ST_OFF;     // a[3], d[3]
    float* Araw  = ws + RAW_OFF;    // 512*676*9
    float* A     = ws + A_OFF;      // 512 x 2028
    float* Bm    = ws + B_OFF;      // 2028 x 128
    float* F     = ws + F_OFF;      // 512 x 128

    conv_raw_stats<<<1352, 256, 0, stream>>>(x, w1, Araw, part);   // single pass over x
    bn_finalize   <<<1,    256, 0, stream>>>(part, gamma, beta, stats);
    conv_combine  <<<1352, 256, 0, stream>>>(Araw, stats, w1, b1, w2, b2, A);
    pack_B        <<<2028, 128, 0, stream>>>(fc1_w, Bm);
    gemm_fc1_wmma <<<32,   256, 0, stream>>>(A, Bm, fc1_b, F);     // 256 waves
    fc2_logsoftmax<<<2,    256, 0, stream>>>(F, fc2_w, fc2_b, out);
}